// Attention_Module_29953101922407
// MI455X (gfx1250) — compile-verified
//
#include <hip/hip_runtime.h>
#include <cmath>

// Problem constants (match reference).
#define BB 32
#define LL 2048
#define TT 1024
#define AA 1024

typedef float v2f __attribute__((ext_vector_type(2)));
typedef float v8f __attribute__((ext_vector_type(8)));

// ---------------------------------------------------------------------------
// Small GEMM via V_WMMA_F32_16X16X4_F32:  out[32,N] = X[32,K] * B + bias
//   WT=1 : Bmat element (k,n) = W[n*K + k]   (contract along W's row)
//   WT=0 : Bmat element (k,n) = W[k*N + n]   (W row-major [K,N])
// One wave per 16x16 output tile. EXEC is all-ones (32 threads, no divergence)
// as required for WMMA.
// ---------------------------------------------------------------------------
template <int WT, int HASBIAS>
__global__ __launch_bounds__(32) void gemm32_wmma(
    const float* __restrict__ X, const float* __restrict__ W,
    const float* __restrict__ bias, float* __restrict__ out,
    int K, int N, float scale) {
  const int lane = threadIdx.x;        // 0..31
  const int lo16 = lane & 15;
  const int hi   = lane >> 4;          // 0 or 1
  const int mtile = blockIdx.y * 16;
  const int n = blockIdx.x * 16 + lo16;
  const int m = mtile + lo16;
  const float* __restrict__ xrow = X + (size_t)m * K;

  v8f acc = {};
  for (int k = 0; k < K; k += 4) {
    const int k0 = k + 2 * hi;
    v2f a, b;
    // A 16x4: VGPR0 = K {0 | 2}, VGPR1 = K {1 | 3} per lane half (ISA 7.12.2)
    a.x = xrow[k0];
    a.y = xrow[k0 + 1];
    if (WT) {
      b.x = W[(size_t)n * K + k0];
      b.y = W[(size_t)n * K + k0 + 1];
    } else {
      b.x = W[(size_t)k0 * N + n];
      b.y = W[(size_t)(k0 + 1) * N + n];
    }
    acc = __builtin_amdgcn_wmma_f32_16x16x4_f32(
        /*neg_a=*/false, a, /*neg_b=*/false, b,
        /*c_mod=*/(short)0, acc, /*reuse_a=*/false, /*reuse_b=*/false);
  }
  // D 16x16 f32: lane -> col (lane%16); VGPR v -> row mtile + 8*hi + v
#pragma unroll
  for (int v = 0; v < 8; ++v) {
    const int row = mtile + 8 * hi + v;
    float val = acc[v];
    if (HASBIAS) val += bias[n];
    out[(size_t)row * N + n] = val * scale;
  }
}

// ---------------------------------------------------------------------------
// c[b] = scale * dot(Q[b,:], bk)
// ---------------------------------------------------------------------------
__device__ inline float waveReduceSum(float v) {
  v += __shfl_xor(v, 16, 32);
  v += __shfl_xor(v, 8, 32);
  v += __shfl_xor(v, 4, 32);
  v += __shfl_xor(v, 2, 32);
  v += __shfl_xor(v, 1, 32);
  return v;
}

__global__ __launch_bounds__(256) void dot_bias(
    const float* __restrict__ Q, const float* __restrict__ bk,
    float* __restrict__ cv, int n, float scale) {
  const int b = blockIdx.x;
  const int tid = threadIdx.x;
  float s = 0.f;
  for (int i = tid; i < n; i += 256) s += Q[(size_t)b * n + i] * bk[i];
  s = waveReduceSum(s);
  __shared__ float sm[8];
  if ((tid & 31) == 0) sm[tid >> 5] = s;
  __syncthreads();
  if (tid == 0) {
    float t = 0.f;
#pragma unroll
    for (int w = 0; w < 8; ++w) t += sm[w];
    cv[b] = t * scale;
  }
}

// ---------------------------------------------------------------------------
// Fused single pass over reviewer_emb (the only big tensor, read once):
//   e      = qk[b,:] . rev[b,l,:] + c[b]        (qk & c pre-scaled by 1/sqrt(A))
//   acc[t] += e * rev[b,l,t]
// Block = 256 threads, each owns 4 consecutive t (float4 / b128 traffic).
// Grid = (chunks, B); partials reduced deterministically by reduce_partial.
// ---------------------------------------------------------------------------
__global__ __launch_bounds__(256) void fused_ew(
    const float* __restrict__ rev, const float* __restrict__ qk,
    const float* __restrict__ cvec, float* __restrict__ partial,
    int Ldim, int Tdim, int rowsPerChunk) {
  const int b = blockIdx.y;
  const int chunk = blockIdx.x;
  const int tid = threadIdx.x;
  const int wid = tid >> 5;
  const int lane = tid & 31;
  const int T4 = Tdim >> 2;

  const float4* __restrict__ revB =
      (const float4*)(rev + (size_t)b * Ldim * Tdim);
  const float4 q4 = *(const float4*)(qk + (size_t)b * Tdim + tid * 4);
  const float cb = cvec[b];

  float4 acc = make_float4(0.f, 0.f, 0.f, 0.f);
  __shared__ float sred[4][8];

  const int l0 = chunk * rowsPerChunk;
  const int lend = l0 + rowsPerChunk;
  for (int l = l0; l < lend; l += 4) {
    float4 r0 = revB[(size_t)(l + 0) * T4 + tid];
    float4 r1 = revB[(size_t)(l + 1) * T4 + tid];
    float4 r2 = revB[(size_t)(l + 2) * T4 + tid];
    float4 r3 = revB[(size_t)(l + 3) * T4 + tid];
    if (l + 4 < lend)  // speculative prefetch of next row group -> global_prefetch_b8
      __builtin_prefetch(&revB[(size_t)(l + 4) * T4 + tid], 0, 0);

    float p0 = q4.x * r0.x + q4.y * r0.y + q4.z * r0.z + q4.w * r0.w;
    float p1 = q4.x * r1.x + q4.y * r1.y + q4.z * r1.z + q4.w * r1.w;
    float p2 = q4.x * r2.x + q4.y * r2.y + q4.z * r2.z + q4.w * r2.w;
    float p3 = q4.x * r3.x + q4.y * r3.y + q4.z * r3.z + q4.w * r3.w;
    p0 = waveReduceSum(p0);
    p1 = waveReduceSum(p1);
    p2 = waveReduceSum(p2);
    p3 = waveReduceSum(p3);
    if (lane == 0) {
      sred[0][wid] = p0; sred[1][wid] = p1;
      sred[2][wid] = p2; sred[3][wid] = p3;
    }
    __syncthreads();
    float e0 = cb, e1 = cb, e2 = cb, e3 = cb;
#pragma unroll
    for (int w = 0; w < 8; ++w) {
      e0 += sred[0][w]; e1 += sred[1][w];
      e2 += sred[2][w]; e3 += sred[3][w];
    }
    __syncthreads();
    acc.x += e0 * r0.x + e1 * r1.x + e2 * r2.x + e3 * r3.x;
    acc.y += e0 * r0.y + e1 * r1.y + e2 * r2.y + e3 * r3.y;
    acc.z += e0 * r0.z + e1 * r1.z + e2 * r2.z + e3 * r3.z;
    acc.w += e0 * r0.w + e1 * r1.w + e2 * r2.w + e3 * r3.w;
  }
  float4* p = (float4*)(partial +
                        ((size_t)(chunk * gridDim.y + b)) * Tdim + tid * 4);
  *p = acc;
}

__global__ __launch_bounds__(256) void reduce_partial(
    const float* __restrict__ partial, float* __restrict__ out,
    int chunks, int n) {
  const int i = blockIdx.x * 256 + threadIdx.x;
  if (i >= n) return;
  float s = 0.f;
  for (int c = 0; c < chunks; ++c) s += partial[(size_t)c * n + i];
  out[i] = s;
}

// ---------------------------------------------------------------------------
extern "C" void kernel_launch(void* const* d_in, const int* in_sizes, int n_in,
                              void* d_out, int out_size, void* d_ws,
                              size_t ws_size, hipStream_t stream) {
  const float* sub = (const float*)d_in[0];  // [B, T]
  const float* rev = (const float*)d_in[1];  // [B, L, T]
  const float* Wq  = (const float*)d_in[2];  // [A, T]
  const float* bq  = (const float*)d_in[3];  // [A]
  const float* Wk  = (const float*)d_in[4];  // [A, T]
  const float* bk  = (const float*)d_in[5];  // [A]
  float* out = (float*)d_out;                // [B, T]
  float* ws  = (float*)d_ws;

  const float invsA = 1.0f / sqrtf((float)AA);

  // Workspace layout (floats)
  float* Q   = ws;                       // B*A
  float* qk  = Q + BB * AA;              // B*T
  float* cv  = qk + BB * TT;             // B (padded to 64)
  float* par = cv + 64;                  // LC*B*T

  // Pick largest chunk count that fits in the workspace (deterministic).
  int LC = 16;
  const size_t baseFloats = (size_t)BB * AA + (size_t)BB * TT + 64;
  while (LC > 1 &&
         (baseFloats + (size_t)LC * BB * TT) * sizeof(float) > ws_size)
    LC >>= 1;

  // 1) Q[b,a] = sum_t sub[b,t]*Wq[a,t] + bq[a]           (WMMA f32 16x16x4)
  gemm32_wmma<1, 1><<<dim3(AA / 16, BB / 16), 32, 0, stream>>>(
      sub, Wq, bq, Q, /*K=*/TT, /*N=*/AA, 1.0f);
  // 2) qk[b,t] = (sum_a Q[b,a]*Wk[a,t]) / sqrt(A)        (WMMA f32 16x16x4)
  gemm32_wmma<0, 0><<<dim3(TT / 16, BB / 16), 32, 0, stream>>>(
      Q, Wk, nullptr, qk, /*K=*/AA, /*N=*/TT, invsA);
  // 3) c[b] = (Q[b,:] . bk) / sqrt(A)
  dot_bias<<<BB, 256, 0, stream>>>(Q, bk, cv, AA, invsA);
  // 4) Fused single HBM pass over reviewer_emb
  fused_ew<<<dim3(LC, BB), 256, 0, stream>>>(rev, qk, cv, par, LL, TT,
                                             LL / LC);
  // 5) Deterministic partial reduction -> ww[B,T]
  reduce_partial<<<(BB * TT) / 256, 256, 0, stream>>>(par, out, LC, BB * TT);
}